// NCCLoss_85555748536888
// MI455X (gfx1250) — compile-verified
//
#include <hip/hip_runtime.h>
#include <stdint.h>

// Problem constants (fixed by the reference).
#define DIM      192
#define BATCH    2
#define TX       64             // tile width (outputs) per block
#define TY       16             // tile height (outputs) per block
#define HX       18             // float4 chunks per halo row: (TX + 8) / 4
#define HY       18             // halo rows: TY + 2
#define ZCHUNK   24
#define NZC      (DIM / ZCHUNK)    // 8
#define NBX      (DIM / TX)        // 3
#define NBY      (DIM / TY)        // 12
#define SMOOTH_DR 1e-5f

// ---- CDNA5 async global->LDS DMA (ASYNCcnt path, gfx1250) ------------------
// GVS mode: mem_addr = SGPR64 + VGPR32. First operand is the LDS byte-address
// VGPR (low 32 bits of the flat shared pointer == LDS offset).
__device__ __forceinline__ void async_g2l_b128(const float* sbase, uint32_t voff,
                                               const float4* lds_dst) {
    uint32_t lds_addr = (uint32_t)(uintptr_t)lds_dst;
    asm volatile("global_load_async_to_lds_b128 %0, %1, %2"
                 :: "v"(lds_addr), "v"(voff), "s"(sbase)
                 : "memory");
}
__device__ __forceinline__ void wait_async0() {
    asm volatile("s_wait_asynccnt 0" ::: "memory");
}

__global__ __launch_bounds__(256)
void lncc_partial_kernel(const float* __restrict__ pred,
                         const float* __restrict__ label,
                         float* __restrict__ partial) {
    // Double-buffered raw halo planes, held as 16B chunks (async b128 targets).
    __shared__ float4 rawT[2][HY][HX];      // t = label
    __shared__ float4 rawP[2][HY][HX];      // p = pred
    __shared__ float4 rs4[5][HY][TX / 4];   // x-direction 3-tap sums (5 quantities)
    __shared__ float  red[256];

    const int tid = threadIdx.x;
    const int c   = tid & 15;               // float4 output column (0..15)
    const int ty  = tid >> 4;               // output row (0..15)

    const int bx = blockIdx.x;              // 0..2
    const int by = blockIdx.y;              // 0..11
    const int bz = blockIdx.z;              // 0..BATCH*NZC-1
    const int b  = bz / NZC;
    const int z0 = (bz % NZC) * ZCHUNK;
    const int x0 = bx * TX;
    const int y0 = by * TY;

    const size_t img = (size_t)b * DIM * DIM * DIM;
    const float* tvol = label + img;
    const float* pvol = pred  + img;

    // Zero raw buffers once: out-of-range halo chunks are never overwritten,
    // so they provide the zero padding for the whole chunk.
    for (int i = tid; i < 2 * HY * HX; i += 256) {
        (&rawT[0][0][0])[i] = make_float4(0.f, 0.f, 0.f, 0.f);
        (&rawP[0][0][0])[i] = make_float4(0.f, 0.f, 0.f, 0.f);
    }
    __syncthreads();

    // Async-copy plane zp into raw slot (zp & 1) as aligned b128 chunks.
    // Chunk grid is 4-float aligned in gx and image bounds are 4-aligned, so
    // every chunk is fully in range or fully out (out -> stays zero).
    auto issue_plane = [&](int zp) {
        if (zp < 0 || zp >= DIM) return;
        const int slot = zp & 1;
        const float* tpl = tvol + (size_t)zp * DIM * DIM;
        const float* ppl = pvol + (size_t)zp * DIM * DIM;
        for (int i = tid; i < HY * HX; i += 256) {
            const int yy  = i / HX;
            const int k   = i - yy * HX;
            const int gy  = y0 + yy - 1;
            const int gx0 = x0 - 4 + 4 * k;
            if (gy >= 0 && gy < DIM && gx0 >= 0 && gx0 < DIM) {
                const uint32_t voff = (uint32_t)((gy * DIM + gx0) * 4);
                async_g2l_b128(tpl, voff, &rawT[slot][yy][k]);
                async_g2l_b128(ppl, voff, &rawP[slot][yy][k]);
            }
        }
    };

    // Register ring of per-plane 2D 3x3 sums: 5 quantities x 4 x-outputs.
    float sP2[5][4], sP1[5][4], sC[5][4];
#pragma unroll
    for (int q = 0; q < 5; ++q)
#pragma unroll
        for (int k = 0; k < 4; ++k) { sP2[q][k] = 0.f; sP1[q][k] = 0.f; sC[q][k] = 0.f; }

    float acc = 0.0f;

    issue_plane(z0 - 1);
    for (int zp = z0 - 1; zp <= z0 + ZCHUNK; ++zp) {
        wait_async0();
        __syncthreads();                        // plane zp resident in LDS
        if (zp + 1 <= z0 + ZCHUNK) issue_plane(zp + 1);   // overlap next DMA

        const int  slot   = zp & 1;
        const bool zvalid = (zp >= 0) && (zp < DIM);

        // Stage A: x-direction 3-tap sums for 4 outputs per position.
        // Output x = 4*cc+k uses raw floats L = 4*cc+3+k .. 4*cc+5+k
        // (L = gx - (x0-4)).
        for (int i = tid; i < HY * 16; i += 256) {
            const int yy = i >> 4;
            const int cc = i & 15;
            float rt[4], rp[4], qt[4], qp[4], qx[4];
            if (zvalid) {
                const float* rowT = (const float*)&rawT[slot][yy][0];
                const float* rowP = (const float*)&rawP[slot][yy][0];
                float ta[6], pa[6];
#pragma unroll
                for (int j = 0; j < 6; ++j) {
                    ta[j] = rowT[4 * cc + 3 + j];
                    pa[j] = rowP[4 * cc + 3 + j];
                }
#pragma unroll
                for (int k = 0; k < 4; ++k) {
                    const float t0 = ta[k], t1 = ta[k + 1], t2 = ta[k + 2];
                    const float p0 = pa[k], p1 = pa[k + 1], p2 = pa[k + 2];
                    rt[k] = t0 + t1 + t2;
                    rp[k] = p0 + p1 + p2;
                    qt[k] = fmaf(t2, t2, fmaf(t1, t1, t0 * t0));
                    qp[k] = fmaf(p2, p2, fmaf(p1, p1, p0 * p0));
                    qx[k] = fmaf(t2, p2, fmaf(t1, p1, t0 * p0));
                }
            } else {
#pragma unroll
                for (int k = 0; k < 4; ++k) { rt[k] = rp[k] = qt[k] = qp[k] = qx[k] = 0.f; }
            }
            rs4[0][yy][cc] = make_float4(rt[0], rt[1], rt[2], rt[3]);
            rs4[1][yy][cc] = make_float4(rp[0], rp[1], rp[2], rp[3]);
            rs4[2][yy][cc] = make_float4(qt[0], qt[1], qt[2], qt[3]);
            rs4[3][yy][cc] = make_float4(qp[0], qp[1], qp[2], qp[3]);
            rs4[4][yy][cc] = make_float4(qx[0], qx[1], qx[2], qx[3]);
        }
        __syncthreads();

        // Stage B: rotate register ring, then fold y (b128 LDS reads -> regs).
#pragma unroll
        for (int q = 0; q < 5; ++q)
#pragma unroll
            for (int k = 0; k < 4; ++k) { sP2[q][k] = sP1[q][k]; sP1[q][k] = sC[q][k]; }
#pragma unroll
        for (int q = 0; q < 5; ++q) {
            const float4 a = rs4[q][ty][c];
            const float4 e = rs4[q][ty + 1][c];
            const float4 d = rs4[q][ty + 2][c];
            sC[q][0] = a.x + e.x + d.x;
            sC[q][1] = a.y + e.y + d.y;
            sC[q][2] = a.z + e.z + d.z;
            sC[q][3] = a.w + e.w + d.w;
        }

        // Stage C: fold z in registers and evaluate NCC for plane zp-1.
        const int zout = zp - 1;
        if (zout >= z0 && zout < z0 + ZCHUNK) {
#pragma unroll
            for (int k = 0; k < 4; ++k) {
                const float tsum = sP2[0][k] + sP1[0][k] + sC[0][k];
                const float psum = sP2[1][k] + sP1[1][k] + sC[1][k];
                const float t2s  = sP2[2][k] + sP1[2][k] + sC[2][k];
                const float p2s  = sP2[3][k] + sP1[3][k] + sC[3][k];
                const float tps  = sP2[4][k] + sP1[4][k] + sC[4][k];
                const float inv_kvol = 1.0f / 27.0f;
                const float tavg  = tsum * inv_kvol;
                const float pavg  = psum * inv_kvol;
                const float cross = tps - pavg * tsum;
                const float tvar  = fmaxf(t2s - tavg * tsum, 0.0f);
                const float pvar  = fmaxf(p2s - pavg * psum, 0.0f);
                acc += (cross * cross) / (tvar * pvar + SMOOTH_DR);
            }
        }
        // next iteration's wait+barrier protects raw/rs reuse
    }

    // Deterministic block reduction.
    red[tid] = acc;
    __syncthreads();
#pragma unroll
    for (int s = 128; s > 0; s >>= 1) {
        if (tid < s) red[tid] += red[tid + s];
        __syncthreads();
    }
    if (tid == 0) {
        const int idx = (bz * NBY + by) * NBX + bx;
        partial[idx] = red[0];
    }
}

__global__ __launch_bounds__(256)
void lncc_finalize_kernel(const float* __restrict__ partial, int n,
                          float* __restrict__ out) {
    __shared__ float red[256];
    float a = 0.0f;
    for (int i = threadIdx.x; i < n; i += 256) a += partial[i];
    red[threadIdx.x] = a;
    __syncthreads();
#pragma unroll
    for (int s = 128; s > 0; s >>= 1) {
        if (threadIdx.x < s) red[threadIdx.x] += red[threadIdx.x + s];
        __syncthreads();
    }
    if (threadIdx.x == 0) {
        const float inv_n = 1.0f / (float)((size_t)BATCH * DIM * DIM * DIM);
        out[0] = -red[0] * inv_n;
    }
}

extern "C" void kernel_launch(void* const* d_in, const int* in_sizes, int n_in,
                              void* d_out, int out_size, void* d_ws, size_t ws_size,
                              hipStream_t stream) {
    (void)in_sizes; (void)n_in; (void)out_size; (void)ws_size;
    const float* pred  = (const float*)d_in[0];
    const float* label = (const float*)d_in[1];
    float* partial = (float*)d_ws;
    float* out     = (float*)d_out;

    const int nblocks = NBX * NBY * BATCH * NZC;   // 576
    dim3 grid(NBX, NBY, BATCH * NZC);
    lncc_partial_kernel<<<grid, 256, 0, stream>>>(pred, label, partial);
    lncc_finalize_kernel<<<1, 256, 0, stream>>>(partial, nblocks, out);
}